// hpNet_base_6614249635913
// MI455X (gfx1250) — compile-verified
//
#include <hip/hip_runtime.h>
#include <hip/hip_bf16.h>
#include <math.h>

// ---------------------------------------------------------------------------
// Hyperbolic memory-attention net on MI455X (gfx1250, wave32).
// Heavy op: 65536 x (64x64 matvec). Only 32 distinct relation matrices
// (512KB, L2-resident) -> bucket memories by relation, run grouped GEMM
// with V_WMMA_F32_16X16X4_F32 (fp32-exact), fuse proj_tan0_exp epilogue.
// ---------------------------------------------------------------------------

typedef float v2f __attribute__((ext_vector_type(2)));
typedef float v8f __attribute__((ext_vector_type(8)));

#define N_ENTITY 200000
#define N_REL    32
#define DIM      64
#define BATCH    1024
#define HOPS     2
#define MEMS     32
#define NSLOT    (HOPS*BATCH*MEMS)      // 65536 flat [H,B,M] slots
#define MAXTILES (NSLOT/16 + N_REL)     // 4128 upper bound on 16-row tiles
#define MAXNORM  0.996f                 // 1 - 4e-3
#define EPSF     1e-7f
#define MINNRM   1e-15f

__device__ __forceinline__ float redWave(float v) {
  v += __shfl_xor(v, 1, 32);
  v += __shfl_xor(v, 2, 32);
  v += __shfl_xor(v, 4, 32);
  v += __shfl_xor(v, 8, 32);
  v += __shfl_xor(v, 16, 32);
  return v;
}

__device__ __forceinline__ float artanh_c(float x) {
  float c = fminf(fmaxf(x, -1.0f + EPSF), 1.0f - EPSF);
  return 0.5f * (log1pf(c) - log1pf(-c));
}

// --------------------------- bucketing ------------------------------------
__global__ void k_zero(int* counts) {
  int t = threadIdx.x;
  if (t < N_REL) counts[t] = 0;
}

__global__ void k_count(const int* __restrict__ mem_r, int* counts) {
  int i = blockIdx.x * blockDim.x + threadIdx.x;
  if (i < NSLOT) atomicAdd(&counts[mem_r[i]], 1);
}

__global__ void k_scan(const int* counts, int* offsets, int* cursor, int* tileOff) {
  if (blockIdx.x == 0 && threadIdx.x == 0) {
    int acc = 0, tacc = 0;
    for (int r = 0; r < N_REL; ++r) {
      offsets[r] = acc; cursor[r] = acc; tileOff[r] = tacc;
      acc  += counts[r];
      tacc += (counts[r] + 15) >> 4;
    }
    offsets[N_REL] = acc; tileOff[N_REL] = tacc;
  }
}

__global__ void k_scatter(const int* __restrict__ mem_r, int* cursor, int* bucket) {
  int i = blockIdx.x * blockDim.x + threadIdx.x;
  if (i < NSLOT) {
    int r = mem_r[i];
    int pos = atomicAdd(&cursor[r], 1);
    bucket[pos] = i;
  }
}

// --------------------- hyp_item = proj(expmap0(item)) ---------------------
__global__ __launch_bounds__(32) void k_item(const int* __restrict__ items,
                                             const float* __restrict__ ent,
                                             float* __restrict__ hyp) {
  int b = blockIdx.x, l = threadIdx.x;
  int id = items[b];
  float u0 = ent[(size_t)id * DIM + l];
  float u1 = ent[(size_t)id * DIM + l + 32];
  float n = fmaxf(sqrtf(redWave(u0 * u0 + u1 * u1)), MINNRM);
  float s = tanhf(n) / n;
  u0 *= s; u1 *= s;
  float un = fmaxf(sqrtf(redWave(u0 * u0 + u1 * u1)), MINNRM);
  if (un > MAXNORM) { float f = MAXNORM / un; u0 *= f; u1 *= f; }
  hyp[b * DIM + l]      = u0;
  hyp[b * DIM + l + 32] = u1;
}

// ------------- grouped GEMM: Rh = proj_tan0_exp( H_r x R_r^T /10 ) --------
__global__ __launch_bounds__(128) void k_gemm(const int* __restrict__ bucket,
                                              const int* __restrict__ offsets,
                                              const int* __restrict__ tileOff,
                                              const int* __restrict__ mem_h,
                                              const float* __restrict__ ent,
                                              const float* __restrict__ rel,
                                              float* __restrict__ RhOut) {
  __shared__ float Alds[16 * 68];     // 16 slots x 64 dims, pad -> bank-free
  __shared__ float Dlds[16 * 68];
  __shared__ float part[16][8];
  __shared__ float scale_s[16];

  int tile = blockIdx.x;
  if (tile >= tileOff[N_REL]) return;           // uniform per block

  int r = 0;
  #pragma unroll
  for (int i = 0; i < N_REL; ++i)
    if (tile >= tileOff[i]) r = i;              // tileOff[r] <= tile < tileOff[r+1]
  int base = offsets[r] + (tile - tileOff[r]) * 16;
  int nrem = offsets[r + 1] - base;             // 1..16 valid slots in tile

  int tid = threadIdx.x;
  // stage A: 16 gathered head vectors (zeros past nrem keep WMMA benign)
  {
    int sl = tid >> 3, chunk = tid & 7;
    float4 va = make_float4(0.f, 0.f, 0.f, 0.f), vb = va;
    if (sl < nrem) {
      int gs  = bucket[base + sl];
      int hid = mem_h[gs];
      const float4* src = (const float4*)(ent + (size_t)hid * DIM + chunk * 8);
      va = src[0]; vb = src[1];
    }
    *(float4*)&Alds[sl * 68 + chunk * 8]     = va;
    *(float4*)&Alds[sl * 68 + chunk * 8 + 4] = vb;
  }
  __syncthreads();

  int w    = tid >> 5;        // wave -> N-tile (i in [16w,16w+16))
  int lane = tid & 31;
  int mrow = lane & 15;
  int half = lane >> 4;
  const float* Bbase = rel + (size_t)r * (DIM * DIM) + (size_t)(w * 16 + mrow) * DIM;
  __builtin_prefetch(Bbase, 0, 0);              // global_prefetch_b8 into L2

  v8f acc = {0.f, 0.f, 0.f, 0.f, 0.f, 0.f, 0.f, 0.f};
  #pragma unroll
  for (int kk = 0; kk < 16; ++kk) {
    // A frag: row m = lane%16, K-pair (2*half, 2*half+1) within this k-step
    v2f a = *(const v2f*)&Alds[mrow * 68 + kk * 4 + 2 * half];
    // B frag: col n = w*16 + lane%16, same K-pair striping (B[j][i] = R[i][j])
    v2f bv = *(const v2f*)(Bbase + kk * 4 + 2 * half);
    acc = __builtin_amdgcn_wmma_f32_16x16x4_f32(false, a, false, bv,
                                                (short)0, acc, false, false);
  }
  // D layout: vgpr v -> slot v + 8*half, col = lane%16 ; fold the /10 here
  #pragma unroll
  for (int v = 0; v < 8; ++v)
    Dlds[(v + 8 * half) * 68 + w * 16 + mrow] = acc[v] * 0.1f;
  __syncthreads();

  // fused proj_tan0_exp over each 64-dim row
  int slot = tid >> 3, j0 = (tid & 7) * 8;
  float ss = 0.f;
  #pragma unroll
  for (int k = 0; k < 8; ++k) { float x = Dlds[slot * 68 + j0 + k]; ss += x * x; }
  part[slot][tid & 7] = ss;
  __syncthreads();
  if (tid < 16) {
    float nn = 0.f;
    #pragma unroll
    for (int k = 0; k < 8; ++k) nn += part[tid][k];
    float n  = fmaxf(sqrtf(nn), MINNRM);
    float s1 = tanhf(n) / n;                 // expmap0 scale
    float un = fmaxf(tanhf(n), MINNRM);      // |u| after expmap0
    float s2 = (un > MAXNORM) ? (MAXNORM / un) : 1.0f;
    scale_s[tid] = s1 * s2;
  }
  __syncthreads();
  if (slot < nrem) {
    int gs = bucket[base + slot];
    float sc = scale_s[slot];
    float* dst = RhOut + (size_t)gs * DIM + j0;
    #pragma unroll
    for (int k = 0; k < 8; ++k) dst[k] = Dlds[slot * 68 + j0 + k] * sc;
  }
}

// --------- per-b: attention + Klein aggregation over both hops -------------
__global__ __launch_bounds__(256) void k_agg(const int* __restrict__ mem_t,
                                             const float* __restrict__ Rh,
                                             const float* __restrict__ hyp,
                                             const float* __restrict__ ent,
                                             float* __restrict__ out) {
  __shared__ float qs[64];
  __shared__ float kle[64 * 65];
  __shared__ float p1s[64];
  __shared__ float Sh[2];
  __shared__ float kleo[2][64];
  __shared__ float p2s[2];
  __shared__ float ovec[64];
  __shared__ float y2s;

  int b = blockIdx.x, tid = threadIdx.x;
  if (tid < 64) qs[tid] = hyp[b * DIM + tid];
  __syncthreads();

  int w = tid >> 5, lane = tid & 31;
  if (w == 0) {
    float q0 = qs[lane], q1 = qs[lane + 32];
    float y2 = redWave(q0 * q0 + q1 * q1);
    if (lane == 0) y2s = y2;
  }
  __syncthreads();

  float q0 = qs[lane], q1 = qs[lane + 32];
  #pragma unroll 1
  for (int it = 0; it < 8; ++it) {
    int s = w * 8 + it;                    // slot 0..63: h = s>>5, m = s&31
    int h = s >> 5, m = s & 31;
    int gs = (h * BATCH + b) * MEMS + m;   // flat [H,B,M]
    float p0 = Rh[(size_t)gs * DIM + lane];
    float p1 = Rh[(size_t)gs * DIM + lane + 32];
    // sqdist(Rh, hyp_item) via mobius_add(-p, q)
    float x2  = redWave(p0 * p0 + p1 * p1);
    float xy  = -redWave(p0 * q0 + p1 * q1);
    float y2  = y2s;
    float cA  = 1.0f + 2.0f * xy + y2;
    float cB  = 1.0f - x2;
    float den = fmaxf(1.0f + 2.0f * xy + x2 * y2, MINNRM);
    float n0  = (cA * (-p0) + cB * q0) / den;
    float n1  = (cA * (-p1) + cB * q1) / den;
    float dd  = 2.0f * artanh_c(sqrtf(redWave(n0 * n0 + n1 * n1)));
    float att = expf(-0.2f * (dd * dd) - 0.05f);
    // kle_t = to_klein(proj_tan0_exp(t)), lorentz weight
    int ti = mem_t[gs];
    float t0 = ent[(size_t)ti * DIM + lane];
    float t1 = ent[(size_t)ti * DIM + lane + 32];
    float tn = fmaxf(sqrtf(redWave(t0 * t0 + t1 * t1)), MINNRM);
    float sc = tanhf(tn) / tn;
    float u0 = t0 * sc, u1 = t1 * sc;
    float un = fmaxf(sqrtf(redWave(u0 * u0 + u1 * u1)), MINNRM);
    if (un > MAXNORM) { float f = MAXNORM / un; u0 *= f; u1 *= f; }
    float uu = redWave(u0 * u0 + u1 * u1);
    float kf = 2.0f / (1.0f + uu);
    float k0 = u0 * kf, k1 = u1 * kf;
    kle[s * 65 + lane]      = k0;
    kle[s * 65 + lane + 32] = k1;
    float kk2 = redWave(k0 * k0 + k1 * k1);
    float lf  = rsqrtf(fmaxf(1.0f - kk2, EPSF));
    if (lane == 0) p1s[s] = lf * att;
  }
  __syncthreads();
  if (tid < 2) {
    float s = 0.f;
    for (int m = 0; m < MEMS; ++m) s += p1s[tid * MEMS + m];
    Sh[tid] = fmaxf(s, EPSF);
  }
  __syncthreads();
  if (tid < 128) {
    int h = tid >> 6, d = tid & 63;
    float inv = 1.0f / Sh[h], acc = 0.f;
    for (int m = 0; m < MEMS; ++m) acc += p1s[h * MEMS + m] * kle[(h * MEMS + m) * 65 + d];
    kleo[h][d] = acc * inv;
  }
  __syncthreads();
  if (tid < 2) {
    float ss = 0.f;
    for (int d = 0; d < DIM; ++d) ss += kleo[tid][d] * kleo[tid][d];
    p2s[tid] = rsqrtf(fmaxf(1.0f - ss, EPSF));
  }
  __syncthreads();
  if (tid < 64) {
    float tot = fmaxf(p2s[0] + p2s[1], EPSF);
    ovec[tid] = (p2s[0] * kleo[0][tid] + p2s[1] * kleo[1][tid]) / tot;
  }
  __syncthreads();
  if (w == 0) {
    float o0 = ovec[lane], o1 = ovec[lane + 32];
    float oo = redWave(o0 * o0 + o1 * o1);
    float kt = 1.0f / (1.0f + sqrtf(fmaxf(1.0f - oo, EPSF)));   // klein_to
    float h0 = o0 * kt, h1 = o1 * kt;
    float hn = fmaxf(sqrtf(redWave(h0 * h0 + h1 * h1)), MINNRM);
    if (hn > MAXNORM) { float f = MAXNORM / hn; h0 *= f; h1 *= f; }
    float x2  = redWave(h0 * h0 + h1 * h1);
    float xy  = -redWave(h0 * q0 + h1 * q1);
    float y2  = y2s;
    float cA  = 1.0f + 2.0f * xy + y2;
    float cB  = 1.0f - x2;
    float den = fmaxf(1.0f + 2.0f * xy + x2 * y2, MINNRM);
    float n0  = (cA * (-h0) + cB * q0) / den;
    float n1  = (cA * (-h1) + cB * q1) / den;
    float dd  = 2.0f * artanh_c(sqrtf(redWave(n0 * n0 + n1 * n1)));
    float d2  = dd * dd;
    if (lane == 0)
      out[b] = 1.0f / (expf(fminf(d2 - 2.0f, 40.0f)) + 1.0f);
  }
}

// ---------------------------------------------------------------------------
extern "C" void kernel_launch(void* const* d_in, const int* in_sizes, int n_in,
                              void* d_out, int out_size, void* d_ws, size_t ws_size,
                              hipStream_t stream) {
  const int*   items = (const int*)d_in[1];
  const int*   mem_h = (const int*)d_in[3];
  const int*   mem_r = (const int*)d_in[4];
  const int*   mem_t = (const int*)d_in[5];
  const float* ent   = (const float*)d_in[6];
  const float* rel   = (const float*)d_in[8];
  float*       out   = (float*)d_out;

  // workspace layout
  float* ws_f   = (float*)d_ws;
  float* Rh     = ws_f;                       // 65536*64 f32 = 16 MB
  float* hyp    = ws_f + (size_t)NSLOT * DIM; // 1024*64 f32
  int*   ip     = (int*)(hyp + BATCH * DIM);
  int* counts   = ip;            // 32
  int* offsets  = ip + 32;       // 33
  int* cursor   = ip + 65;       // 32
  int* tileOff  = ip + 97;       // 33
  int* bucket   = ip + 130;      // 65536

  k_zero<<<1, 64, 0, stream>>>(counts);
  k_count<<<NSLOT / 256, 256, 0, stream>>>(mem_r, counts);
  k_scan<<<1, 1, 0, stream>>>(counts, offsets, cursor, tileOff);
  k_scatter<<<NSLOT / 256, 256, 0, stream>>>(mem_r, cursor, bucket);
  k_item<<<BATCH, 32, 0, stream>>>(items, ent, hyp);
  k_gemm<<<MAXTILES, 128, 0, stream>>>(bucket, offsets, tileOff, mem_h, ent, rel, Rh);
  k_agg<<<BATCH, 256, 0, stream>>>(mem_t, Rh, hyp, ent, out);
}